// SteinerTopo_1692217115489
// MI455X (gfx1250) — compile-verified
//
#include <hip/hip_runtime.h>
#include <hip/hip_bf16.h>

// Native clang vector type: accepted by __builtin_nontemporal_store,
// lowers to a single global_store_b64.
typedef float v2f __attribute__((ext_vector_type(2)));

// ---------------------------------------------------------------------------
// Kernel 1: one thread per net. Serial min/max over the net's 2..8 pins.
// Also writes net_branch_start[n] = netpin_start[n] - n  (num_nets+1 entries).
// Deterministic: no atomics, each output written by exactly one lane.
// Output stores are non-temporal (write-once streams) so the 22 MB pos[]
// array stays resident in the 192 MB L2 for the random gathers.
// ---------------------------------------------------------------------------
__global__ void hpwl_per_net_kernel(
    const float* __restrict__ pos,          // 2*num_pins (x then y)
    const int*   __restrict__ flat_netpin,  // num_pins
    const int*   __restrict__ netpin_start, // num_nets+1
    const int*   __restrict__ ignore_deg,   // 1 element
    float* __restrict__ wl,                 // num_nets
    float* __restrict__ net_branch_start,   // num_nets+1
    int num_nets, int num_pins)
{
    const int n = blockIdx.x * blockDim.x + threadIdx.x;
    if (n > num_nets) return;

    const int s = netpin_start[n];
    // net_branch_start = netpin_start - arange(num_nets+1)
    __builtin_nontemporal_store((float)(s - n), &net_branch_start[n]);
    if (n >= num_nets) return;

    const int e = netpin_start[n + 1];

    // Prefetch the next cacheline of the pin-index stream (gfx1250
    // global_prefetch_b8).
    __builtin_prefetch(flat_netpin + e, 0, 1);

    const float* __restrict__ x = pos;
    const float* __restrict__ y = pos + num_pins;

    float xmin =  3.402823e38f, xmax = -3.402823e38f;
    float ymin =  3.402823e38f, ymax = -3.402823e38f;

    for (int p = s; p < e; ++p) {
        const int pin = flat_netpin[p];
        const float px = x[pin];
        const float py = y[pin];
        xmin = fminf(xmin, px);
        xmax = fmaxf(xmax, px);
        ymin = fminf(ymin, py);
        ymax = fmaxf(ymax, py);
    }

    const float hpwl   = (xmax - xmin) + (ymax - ymin);
    const int   degree = e - s;
    __builtin_nontemporal_store((degree < ignore_deg[0]) ? hpwl : 0.0f, &wl[n]);
}

// ---------------------------------------------------------------------------
// Kernel 2: one thread per pin. Streaming writes for nodes / pin_relate_* /
// branch_u / branch_v. nodes written as a 2-wide vector -> global_store_b64;
// all output stores non-temporal (never re-read).
// ---------------------------------------------------------------------------
__global__ void per_pin_kernel(
    const float* __restrict__ pos,          // 2*num_pins
    const int*   __restrict__ flat_netpin,  // num_pins
    const int*   __restrict__ net_ids,      // num_pins
    v2f*   __restrict__ nodes,              // num_pins x (px, py)
    float* __restrict__ pin_relate_x,       // num_pins
    float* __restrict__ pin_relate_y,       // num_pins
    float* __restrict__ branch_u,           // num_pins-1
    float* __restrict__ branch_v,           // num_pins-1
    int num_pins)
{
    const int p = blockIdx.x * blockDim.x + threadIdx.x;
    if (p >= num_pins) return;

    const int   pin = flat_netpin[p];
    const float px  = pos[pin];
    const float py  = pos[pin + num_pins];

    v2f node;
    node.x = px;
    node.y = py;
    __builtin_nontemporal_store(node, &nodes[p]);
    __builtin_nontemporal_store((float)pin, &pin_relate_x[p]);
    __builtin_nontemporal_store((float)(pin + num_pins), &pin_relate_y[p]);

    if (p < num_pins - 1) {
        __builtin_nontemporal_store((float)p, &branch_u[p]);
        const bool same = (net_ids[p + 1] == net_ids[p]);
        __builtin_nontemporal_store((float)(same ? (p + 1) : p), &branch_v[p]);
    }
}

// ---------------------------------------------------------------------------
// Launch. Inputs (setup_inputs order):
//   d_in[0] pos           float32  [2*num_pins]
//   d_in[1] flat_netpin   int32    [num_pins]      (JAX x64 disabled)
//   d_in[2] netpin_start  int32    [num_nets+1]
//   d_in[3] net_ids       int32    [num_pins]
//   d_in[4] ignore_net_degree int32 [1]
// Output (flat float32 concat, return order):
//   wl | nodes | pin_relate_x | pin_relate_y | branch_u | branch_v |
//   net_branch_start
// ---------------------------------------------------------------------------
extern "C" void kernel_launch(void* const* d_in, const int* in_sizes, int n_in,
                              void* d_out, int out_size, void* d_ws, size_t ws_size,
                              hipStream_t stream)
{
    (void)n_in; (void)out_size; (void)d_ws; (void)ws_size;

    const float* pos          = (const float*)d_in[0];
    const int*   flat_netpin  = (const int*)d_in[1];
    const int*   netpin_start = (const int*)d_in[2];
    const int*   net_ids      = (const int*)d_in[3];
    const int*   ignore_deg   = (const int*)d_in[4];

    const int num_pins = in_sizes[1];
    const int num_nets = in_sizes[2] - 1;

    float* out = (float*)d_out;
    float* wl               = out;
    v2f*   nodes            = (v2f*)(wl + (size_t)num_nets);
    float* pin_relate_x     = (float*)nodes + 2 * (size_t)num_pins;
    float* pin_relate_y     = pin_relate_x + (size_t)num_pins;
    float* branch_u         = pin_relate_y + (size_t)num_pins;
    float* branch_v         = branch_u + (size_t)(num_pins - 1);
    float* net_branch_start = branch_v + (size_t)(num_pins - 1);

    const int block = 256;  // 8 wave32 waves

    // Kernel 1 covers num_nets+1 threads (last one only writes
    // net_branch_start's tail element).
    const int grid1 = (num_nets + 1 + block - 1) / block;
    hpwl_per_net_kernel<<<grid1, block, 0, stream>>>(
        pos, flat_netpin, netpin_start, ignore_deg,
        wl, net_branch_start, num_nets, num_pins);

    const int grid2 = (num_pins + block - 1) / block;
    per_pin_kernel<<<grid2, block, 0, stream>>>(
        pos, flat_netpin, net_ids,
        nodes, pin_relate_x, pin_relate_y, branch_u, branch_v, num_pins);
}